// MemStream_14817637171598
// MI455X (gfx1250) — compile-verified
//
#include <hip/hip_runtime.h>
#include <hip/hip_bf16.h>
#include <math.h>

#define IN_DIM   2048
#define OUT_DIM  4096
#define MEM_LEN  32768
#define BETA     5000.0f
#define CHUNKS   64
#define CHUNK_ROWS (MEM_LEN / CHUNKS)   // 512

typedef __attribute__((ext_vector_type(2))) float v2f;
typedef __attribute__((ext_vector_type(8))) float v8f;

// ---------------------------------------------------------------------------
// Kernel 1: per-column partial sum / sumsq of mem_data, fused with the copy of
// mem_data into the output buffer. Consecutive lanes touch consecutive columns
// -> fully coalesced b32 streams in both directions.
// grid = (IN_DIM/256, CHUNKS), block = 256
// ---------------------------------------------------------------------------
__global__ void stats_copy(const float* __restrict__ mem_data,
                           float* __restrict__ out_md,
                           float* __restrict__ psum,
                           float* __restrict__ psq) {
    const int col   = blockIdx.x * 256 + threadIdx.x;
    const int chunk = blockIdx.y;
    const size_t r0 = (size_t)chunk * CHUNK_ROWS;
    float s = 0.0f, q = 0.0f;
    for (int i = 0; i < CHUNK_ROWS; ++i) {
        size_t idx = (r0 + i) * IN_DIM + col;
        float v = mem_data[idx];
        s += v;
        q += v * v;
        out_md[idx] = v;
    }
    psum[chunk * IN_DIM + col] = s;
    psq [chunk * IN_DIM + col] = q;
}

// ---------------------------------------------------------------------------
// Kernel 2: reduce partials -> mean, std (ddof=1); compute normalized query
// new = (x - mean)/std with zero-std guard. Also init loss accumulator to +inf.
// grid = IN_DIM/256, block = 256
// ---------------------------------------------------------------------------
__global__ void finalize_stats(const float* __restrict__ psum,
                               const float* __restrict__ psq,
                               const float* __restrict__ x,
                               float* __restrict__ newv,
                               unsigned int* __restrict__ loss_bits) {
    const int col = blockIdx.x * 256 + threadIdx.x;
    float s = 0.0f, q = 0.0f;
    for (int ch = 0; ch < CHUNKS; ++ch) {
        s += psum[ch * IN_DIM + col];
        q += psq [ch * IN_DIM + col];
    }
    const float N = (float)MEM_LEN;
    float mean = s / N;
    float var  = (q - N * mean * mean) / (N - 1.0f);
    var = fmaxf(var, 0.0f);
    float sd = sqrtf(var);
    newv[col] = (sd == 0.0f) ? 0.0f : (x[col] - mean) / sd;
    if (col == 0) *loss_bits = 0x7f800000u;   // +inf
}

// ---------------------------------------------------------------------------
// Kernel 3: enc = tanh(new @ W_enc + b_enc) via V_WMMA_F32_16X16X4_F32.
// Each wave owns a 16-column tile of enc and chains K=2048 in steps of 4,
// with 4 independent accumulators for ILP. The 1-row query is broadcast to
// all 16 M-rows of the A fragment; row M=0 of the accumulator is the result.
// A layout (16x4 f32): lanes 0-15 hold K={k,k+1}, lanes 16-31 hold K={k+2,k+3}.
// B layout (4x16 f32): VGPR0/1 lanes 0-15 = rows K=k,k+1; lanes 16-31 = K=k+2,k+3.
// grid = OUT_DIM/(16*8) = 32, block = 256 (8 waves)
// ---------------------------------------------------------------------------
__global__ void enc_gemv_wmma(const float* __restrict__ newv,
                              const float* __restrict__ W,   // [IN_DIM, OUT_DIM]
                              const float* __restrict__ b,   // [OUT_DIM]
                              float* __restrict__ enc) {
    const int lane = threadIdx.x & 31;
    const int wave = threadIdx.x >> 5;
    const int n0   = (blockIdx.x * 8 + wave) * 16;
    const int half = lane >> 4;     // 0: lanes 0-15, 1: lanes 16-31
    const int l    = lane & 15;
    const float* Wc = W + n0 + l;   // column base for this lane

    v8f c0 = {}, c1 = {}, c2 = {}, c3 = {};

    for (int k = 0; k < IN_DIM; k += 16) {
#define DO_WMMA(CACC, KB)                                                     \
        {                                                                     \
            int kk = (KB) + 2 * half;                                         \
            v2f a, bm;                                                        \
            a.x  = newv[kk];                                                  \
            a.y  = newv[kk + 1];                                              \
            bm.x = Wc[(size_t)kk * OUT_DIM];                                  \
            bm.y = Wc[(size_t)(kk + 1) * OUT_DIM];                            \
            CACC = __builtin_amdgcn_wmma_f32_16x16x4_f32(                     \
                false, a, false, bm, (short)0, CACC, false, false);           \
        }
        DO_WMMA(c0, k)
        DO_WMMA(c1, k + 4)
        DO_WMMA(c2, k + 8)
        DO_WMMA(c3, k + 12)
#undef DO_WMMA
    }

    // VGPR0 of C/D: lanes 0-15 hold M=0, N=lane -> the GEMV result.
    float r = c0[0] + c1[0] + c2[0] + c3[0];
    if (half == 0) {
        enc[n0 + l] = tanhf(r + b[n0 + l]);
    }
}

// ---------------------------------------------------------------------------
// Kernel 4: dists = sum |memory_row - enc|, fused with the copy of memory to
// the output buffer; deterministic global min via uint atomicMin on float bits
// (all dists are >= 0 so the IEEE ordering matches integer ordering).
// grid = MEM_LEN/8 = 4096, block = 256 (8 waves, one row per wave)
// ---------------------------------------------------------------------------
__global__ void l1_min_copy(const float* __restrict__ memory,
                            const float* __restrict__ enc,
                            float* __restrict__ out_mem,
                            unsigned int* __restrict__ loss_bits) {
    __shared__ float se[OUT_DIM];
    for (int i = threadIdx.x; i < OUT_DIM; i += 256) se[i] = enc[i];
    __syncthreads();

    const int lane = threadIdx.x & 31;
    const int wave = threadIdx.x >> 5;
    const size_t row = (size_t)blockIdx.x * 8 + wave;
    const float* mrow = memory  + row * OUT_DIM;
    float*       orow = out_mem + row * OUT_DIM;

    float s = 0.0f;
    for (int c = lane; c < OUT_DIM; c += 32) {
        float m = mrow[c];
        s += fabsf(m - se[c]);
        orow[c] = m;
    }
    for (int off = 16; off > 0; off >>= 1)
        s += __shfl_xor(s, off, 32);
    if (lane == 0)
        atomicMin(loss_bits, __float_as_uint(s));
}

// ---------------------------------------------------------------------------
// Kernel 5: write scalar loss; conditionally overwrite row `count % MEM_LEN`
// of both output copies with enc / x. grid = 1, block = 256
// ---------------------------------------------------------------------------
__global__ void finalize_out(const unsigned int* __restrict__ loss_bits,
                             const float* __restrict__ enc,
                             const float* __restrict__ x,
                             const int* __restrict__ count,
                             float* __restrict__ d_out) {
    float loss = __uint_as_float(*loss_bits);
    if (threadIdx.x == 0) d_out[0] = loss;
    if (loss <= BETA) {
        int pos = count[0] % MEM_LEN;
        float* out_mem = d_out + 1;
        float* out_md  = d_out + 1 + (size_t)MEM_LEN * OUT_DIM;
        for (int i = threadIdx.x; i < OUT_DIM; i += 256)
            out_mem[(size_t)pos * OUT_DIM + i] = enc[i];
        for (int i = threadIdx.x; i < IN_DIM; i += 256)
            out_md[(size_t)pos * IN_DIM + i] = x[i];
    }
}

// ---------------------------------------------------------------------------
extern "C" void kernel_launch(void* const* d_in, const int* in_sizes, int n_in,
                              void* d_out, int out_size, void* d_ws, size_t ws_size,
                              hipStream_t stream) {
    const float* x        = (const float*)d_in[0];   // [1, IN_DIM]
    const float* memory   = (const float*)d_in[1];   // [MEM_LEN, OUT_DIM]
    const float* mem_data = (const float*)d_in[2];   // [MEM_LEN, IN_DIM]
    const float* W_enc    = (const float*)d_in[3];   // [IN_DIM, OUT_DIM]
    const float* b_enc    = (const float*)d_in[4];   // [OUT_DIM]
    const int*   count    = (const int*)  d_in[5];   // scalar

    float* out     = (float*)d_out;
    float* out_mem = out + 1;                               // new_memory
    float* out_md  = out + 1 + (size_t)MEM_LEN * OUT_DIM;   // new_mem_data

    // workspace layout (floats)
    float* ws   = (float*)d_ws;
    float* psum = ws;                          // CHUNKS * IN_DIM
    float* psq  = psum + CHUNKS * IN_DIM;      // CHUNKS * IN_DIM
    float* newv = psq  + CHUNKS * IN_DIM;      // IN_DIM
    float* enc  = newv + IN_DIM;               // OUT_DIM
    unsigned int* loss_bits = (unsigned int*)(enc + OUT_DIM);

    stats_copy    <<<dim3(IN_DIM / 256, CHUNKS), 256, 0, stream>>>(mem_data, out_md, psum, psq);
    finalize_stats<<<IN_DIM / 256,               256, 0, stream>>>(psum, psq, x, newv, loss_bits);
    enc_gemv_wmma <<<OUT_DIM / (16 * 8),         256, 0, stream>>>(newv, W_enc, b_enc, enc);
    l1_min_copy   <<<MEM_LEN / 8,                256, 0, stream>>>(memory, enc, out_mem, loss_bits);
    finalize_out  <<<1,                          256, 0, stream>>>(loss_bits, enc, x, count, out);
}